// GResBlock_90563680403918
// MI455X (gfx1250) — compile-verified
//
#include <hip/hip_runtime.h>
#include <hip/hip_bf16.h>

#define DIM 128

typedef __attribute__((ext_vector_type(2))) float v2f;
typedef __attribute__((ext_vector_type(8))) float v8f;

// Fused dual GEMM: S = X @ W, ACC = X @ WL + bias  (X read once).
// One wave -> 16 rows x 128 cols, via V_WMMA_F32_16X16X4_F32, K=128 in 32 steps.
// 4 waves per block -> 64 rows per block.
__global__ __launch_bounds__(128) void gconv_gemm_dual(
    const float* __restrict__ X,    // [N, 128]
    const float* __restrict__ W,    // [128, 128] row-major (k, n)
    const float* __restrict__ WL,   // [128, 128]
    const float* __restrict__ bias, // [128]
    float* __restrict__ S,          // [N, 128] support out
    float* __restrict__ ACC,        // [N, 128] base out (atomic target later)
    int N)
{
    const int wave    = threadIdx.x >> 5;
    const int lane    = threadIdx.x & 31;
    const int l16     = lane & 15;
    const int halfSel = lane >> 4;               // 0: lanes 0-15, 1: lanes 16-31
    const int rowBase = (blockIdx.x * 4 + wave) * 16;

    v8f accW[8];
    v8f accL[8];
#pragma unroll
    for (int t = 0; t < 8; ++t) { accW[t] = (v8f)0.0f; accL[t] = (v8f)0.0f; }

    // A fragment row for this lane (M = rowBase + l16). Clamp instead of
    // predicating: a wrong-but-valid row only corrupts output row M, and all
    // stores are guarded by (m < N) below. Keeps the K-loop branch-free.
    int arow = rowBase + l16;
    if (arow >= N) arow = N - 1;
    const float* xrow = X + (size_t)arow * DIM;

    for (int k0 = 0; k0 < DIM; k0 += 4) {
        // A 16x4 f32 layout: VGPR0 = K=k0 (lanes 0-15) / K=k0+2 (lanes 16-31),
        //                    VGPR1 = K=k0+1 / K=k0+3
        const int kk = k0 + halfSel * 2;
        v2f a;
        a.x = xrow[kk];
        a.y = xrow[kk + 1];

#pragma unroll
        for (int t = 0; t < 8; ++t) {
            // B 4x16 f32 layout (mirror of A, lane -> N): rows kk, kk+1 of W
            const int col = t * 16 + l16;
            v2f bw, bl;
            bw.x = W [(size_t)kk * DIM + col];
            bw.y = W [(size_t)(kk + 1) * DIM + col];
            bl.x = WL[(size_t)kk * DIM + col];
            bl.y = WL[(size_t)(kk + 1) * DIM + col];

            accW[t] = __builtin_amdgcn_wmma_f32_16x16x4_f32(
                false, a, false, bw, (short)0, accW[t], false, false);
            accL[t] = __builtin_amdgcn_wmma_f32_16x16x4_f32(
                false, a, false, bl, (short)0, accL[t], false, false);
        }
    }

    // Epilogue. C/D layout: VGPR r -> M = rowBase + r + halfSel*8, N = colBase + l16
#pragma unroll
    for (int t = 0; t < 8; ++t) {
        const int   col = t * 16 + l16;
        const float bv  = bias[col];
#pragma unroll
        for (int r = 0; r < 8; ++r) {
            const int m = rowBase + halfSel * 8 + r;
            if (m < N) {
                S  [(size_t)m * DIM + col] = accW[t][r];
                ACC[(size_t)m * DIM + col] = accL[t][r] + bv;
            }
        }
    }
}

// One wave per edge: coalesced 512B gather of support[src], scale by edge
// weight, 4 x global_atomic_add_f32 per lane into ACC[dst]. L2-resident.
__global__ __launch_bounds__(256) void gconv_scatter(
    const float* __restrict__ S,
    const int*   __restrict__ src,
    const int*   __restrict__ dst,
    const float* __restrict__ ew,
    float* __restrict__ ACC,
    int E)
{
    const int e    = blockIdx.x * (blockDim.x >> 5) + (threadIdx.x >> 5);
    const int lane = threadIdx.x & 31;
    if (e >= E) return;

    const int   s = src[e];
    const int   d = dst[e];
    const float w = ew[e];

    const float4 v = ((const float4*)(S + (size_t)s * DIM))[lane];
    float* dp = ACC + (size_t)d * DIM + lane * 4;
    atomicAdd(dp + 0, v.x * w);
    atomicAdd(dp + 1, v.y * w);
    atomicAdd(dp + 2, v.z * w);
    atomicAdd(dp + 3, v.w * w);
}

// out = (x + out) * 0.5, vectorized float4, in place on d_out.
__global__ __launch_bounds__(256) void gconv_finalize(
    const float4* __restrict__ x, float4* __restrict__ out, int n4)
{
    const int i = blockIdx.x * blockDim.x + threadIdx.x;
    if (i >= n4) return;
    const float4 a = x[i];
    const float4 b = out[i];
    float4 r;
    r.x = (a.x + b.x) * 0.5f;
    r.y = (a.y + b.y) * 0.5f;
    r.z = (a.z + b.z) * 0.5f;
    r.w = (a.w + b.w) * 0.5f;
    out[i] = r;
}

extern "C" void kernel_launch(void* const* d_in, const int* in_sizes, int n_in,
                              void* d_out, int out_size, void* d_ws, size_t ws_size,
                              hipStream_t stream) {
    const float* x   = (const float*)d_in[0];   // [N,128]
    const int*   ei  = (const int*)  d_in[1];   // [2,E]
    const float* ew  = (const float*)d_in[2];   // [E]
    const float* W1  = (const float*)d_in[3];
    const float* WL1 = (const float*)d_in[4];
    const float* b1  = (const float*)d_in[5];
    const float* W2  = (const float*)d_in[6];
    const float* WL2 = (const float*)d_in[7];
    const float* b2  = (const float*)d_in[8];

    const int N = in_sizes[0] / DIM;
    const int E = in_sizes[2];
    const int* src = ei;
    const int* dst = ei + E;

    float* S   = (float*)d_ws;                 // [N,128] support
    float* H1  = S + (size_t)N * DIM;          // [N,128] layer-1 result
    float* out = (float*)d_out;                // layer-2 accumulator + output

    const int gemmBlocks    = (N + 63) / 64;
    const int scatterBlocks = (E + 7) / 8;     // 8 waves (edges) per 256-thr block
    const int n4            = (N * DIM) / 4;
    const int finBlocks     = (n4 + 255) / 256;

    // Layer 1: S = x@W1 ; H1 = x@WL1 + b1 ; H1 += scatter(ew * S[src] -> dst)
    gconv_gemm_dual<<<gemmBlocks, 128, 0, stream>>>(x, W1, WL1, b1, S, H1, N);
    gconv_scatter<<<scatterBlocks, 256, 0, stream>>>(S, src, dst, ew, H1, E);

    // Layer 2: S = H1@W2 ; out = H1@WL2 + b2 ; out += scatter
    gconv_gemm_dual<<<gemmBlocks, 128, 0, stream>>>(H1, W2, WL2, b2, S, out, N);
    gconv_scatter<<<scatterBlocks, 256, 0, stream>>>(S, src, dst, ew, out, E);

    // out = (x + out) * 0.5
    gconv_finalize<<<finBlocks, 256, 0, stream>>>((const float4*)x, (float4*)out, n4);
}